// WhiteboxGATBlock_85177791414416
// MI455X (gfx1250) — compile-verified
//
#include <hip/hip_runtime.h>
#include <math.h>
#include <stdint.h>

typedef float v2f __attribute__((ext_vector_type(2)));
typedef float v8f __attribute__((ext_vector_type(8)));
typedef unsigned int v4u __attribute__((ext_vector_type(4)));
typedef int v4i __attribute__((ext_vector_type(4)));
typedef int v8i __attribute__((ext_vector_type(8)));

#define D_DIM 256
#define KH 4
#define RDIM 16

#if defined(__has_builtin)
#if __has_builtin(__builtin_amdgcn_tensor_load_to_lds) && \
    __has_builtin(__builtin_amdgcn_s_wait_tensorcnt)
#define HAVE_TDM 1
#endif
#endif
#ifndef HAVE_TDM
#define HAVE_TDM 0
#endif

// ---------------------------------------------------------------------------
// WMMA helper: accumulate C += A(16xK) * B(Kx16) with V_WMMA_F32_16X16X4_F32.
// A in LDS, row-major, padded stride lda (cols+4 -> bank = (4*row+k)%64,
// conflict-free). B global row-major (ldb), pre-offset to the 16-col tile.
// f32 A-fragment (ISA 7.12.2): lanes 0-15 hold (k0,k0+1) of row=lane,
// lanes 16-31 hold (k0+2,k0+3) of row=lane-16. B mirrors C's half-split.
// ---------------------------------------------------------------------------
__device__ __forceinline__ v8f wmma_k_loop(const float* aLds, int lda,
                                           const float* __restrict__ B, int ldb,
                                           int kLen, v8f acc) {
  const int lane = threadIdx.x & 31;
  const int row  = lane & 15;
  const int koff = (lane >> 4) << 1;               // 0 or 2
  const float* aRow = aLds + row * lda + koff;
  const float* bPtr = B + ((lane >> 4) << 1) * ldb + (lane & 15);
  for (int k0 = 0; k0 < kLen; k0 += 4) {
    v2f a, b;
    a.x = aRow[k0];
    a.y = aRow[k0 + 1];
    b.x = bPtr[(size_t)k0 * ldb];
    b.y = bPtr[(size_t)(k0 + 1) * ldb];
    acc = __builtin_amdgcn_wmma_f32_16x16x4_f32(false, a, false, b,
                                                (short)0, acc, false, false);
  }
  return acc;
}

// ---------------------------------------------------------------------------
// Stage a 16-row x cols f32 tile from global into LDS with row stride
// cols+padDwords. TDM path: wave 0 issues one tensor_load_to_lds with
//   tensor = (cols wide, rowsAvail high, row stride ldg), tile = cols x 16,
//   pad_enable -> pad `padDwords` after every 2^(padCode+1) dwords (== cols),
//   OOB rows (N tail) read as zero per ISA 08 §8.2.
// Fallback: manual float4 staging. Caller must __syncthreads() after.
// ---------------------------------------------------------------------------
#if HAVE_TDM
__device__ __forceinline__ void tdm_stage_tile16(const float* __restrict__ gsrc,
                                                 int ldg, int rowsAvail, int cols,
                                                 float* ldsDst, int padCode,
                                                 int padDwords) {
  if (threadIdx.x < 32) {
    unsigned long long ga = (unsigned long long)(uintptr_t)gsrc;
    unsigned int ldsOff = (unsigned int)(uintptr_t)ldsDst;  // addr[31:0] = LDS offset
    v4u g0;
    g0[0] = 1u;                                              // count=1, gather off
    g0[1] = ldsOff;                                          // lds_addr
    g0[2] = (unsigned int)(ga & 0xffffffffull);              // global_addr[31:0]
    g0[3] = (unsigned int)((ga >> 32) & 0x01ffffffull)       // global_addr[56:32]
          | (2u << 30);                                      // type = 2 (image)
    unsigned int td0 = (unsigned int)cols;
    unsigned int td1 = (unsigned int)(rowsAvail > 0 ? rowsAvail : 0);
    unsigned int d0 = (2u << 16);                            // data_size = 4 bytes
    if (padDwords > 0)
      d0 |= (1u << 20)                                       // pad_enable
          | ((unsigned int)padCode << 22)                    // pad_interval code
          | ((unsigned int)(padDwords - 1) << 25);           // pad_amount-1
    v8i g1;
    g1[0] = (int)d0;
    g1[1] = (int)((td0 & 0xffffu) << 16);                    // tensor_dim0[15:0]
    g1[2] = (int)(((td0 >> 16) & 0xffffu) | ((td1 & 0xffffu) << 16));
    g1[3] = (int)(((td1 >> 16) & 0xffffu) | ((td0 & 0xffffu) << 16)); // tile_dim0=cols
    g1[4] = 16;                                              // tile_dim1 = 16 rows
    g1[5] = (int)(unsigned int)ldg;                          // tensor_dim0_stride[31:0]
    g1[6] = 0;
    g1[7] = 0;
    v4i z4 = {0, 0, 0, 0};
#if __clang_major__ >= 23
    v8i z8 = {0, 0, 0, 0, 0, 0, 0, 0};
    __builtin_amdgcn_tensor_load_to_lds(g0, g1, z4, z4, z8, 0);
#else
    __builtin_amdgcn_tensor_load_to_lds(g0, g1, z4, z4, 0);
#endif
    __builtin_amdgcn_s_wait_tensorcnt((short)0);
  }
}
#endif

__device__ __forceinline__ void stage_tile16(const float* __restrict__ src, int ldg,
                                             int rowBase, int nRows, int cols,
                                             float* lds, int padCode, int padDwords) {
#if HAVE_TDM
  tdm_stage_tile16(src + (size_t)rowBase * ldg, ldg, nRows - rowBase, cols, lds,
                   padCode, padDwords);
#else
  const int lda = cols + padDwords;
  const int q4 = cols >> 2;
  for (int i = threadIdx.x; i < 16 * q4; i += blockDim.x) {
    int r = i / q4;
    int c = (i - r * q4) << 2;
    float4 v = {0.f, 0.f, 0.f, 0.f};
    int row = rowBase + r;
    if (row < nRows) v = *(const float4*)(src + (size_t)row * ldg + c);
    float* p = lds + r * lda + c;
    p[0] = v.x; p[1] = v.y; p[2] = v.z; p[3] = v.w;
  }
#endif
}

// ---------------------------------------------------------------------------
// pack: DmatT[j][i] = Dmat[i][j]; Ucat[d][k*16+r] = U[k][d][r];
//       Uw[k*16+r][d] = softmax(head_w)[k] * U[k][d][r]
// ---------------------------------------------------------------------------
__global__ void pack_kernel(const float* __restrict__ U, const float* __restrict__ Dmat,
                            const float* __restrict__ head_w,
                            float* __restrict__ Ucat, float* __restrict__ Uw,
                            float* __restrict__ DmatT) {
  int idx = blockIdx.x * blockDim.x + threadIdx.x;
  if (idx < D_DIM * D_DIM) {
    int i = idx >> 8, j = idx & 255;
    DmatT[j * D_DIM + i] = Dmat[i * D_DIM + j];
  }
  if (idx < KH * D_DIM * RDIM) {
    int k = idx >> 12;
    int d = (idx >> 4) & 255;
    int r = idx & 15;
    float h0 = head_w[0], h1 = head_w[1], h2 = head_w[2], h3 = head_w[3];
    float m  = fmaxf(fmaxf(h0, h1), fmaxf(h2, h3));
    float e0 = __expf(h0 - m), e1 = __expf(h1 - m);
    float e2 = __expf(h2 - m), e3 = __expf(h3 - m);
    float s  = e0 + e1 + e2 + e3;
    float wk = (k == 0 ? e0 : k == 1 ? e1 : k == 2 ? e2 : e3) / s;
    float u  = U[idx];
    Ucat[d * (KH * RDIM) + k * RDIM + r] = u;
    Uw[(k * RDIM + r) * D_DIM + d] = wk * u;
  }
}

// init per launch (harness does not re-poison between replays)
__global__ void init_kernel(float* __restrict__ segmax, float* __restrict__ segsum,
                            float* __restrict__ agg, int N) {
  int stride = gridDim.x * blockDim.x;
  for (int i = blockIdx.x * blockDim.x + threadIdx.x; i < N * 64; i += stride) {
    agg[i] = 0.f;
    if (i < N * KH) { segmax[i] = -__builtin_inff(); segsum[i] = 0.f; }
  }
}

// ---------------------------------------------------------------------------
// ZU = Z (Nx256) @ Ucat (256x64). One block = 16 rows, 4 waves x one 16-col tile.
// ---------------------------------------------------------------------------
__global__ void zu_gemm_kernel(const float* __restrict__ Z, const float* __restrict__ Ucat,
                               float* __restrict__ ZU, int N) {
  __shared__ float aTile[16 * 260];
  const int rowBase = blockIdx.x * 16;
  stage_tile16(Z, D_DIM, rowBase, N, D_DIM, aTile, /*padCode 256dw*/7, 4);
  __syncthreads();
  const int wave = threadIdx.x >> 5;               // 0..3
  const int colBase = wave * 16;
  v8f acc = {};
  acc = wmma_k_loop(aTile, 260, Ucat + colBase, KH * RDIM, D_DIM, acc);
  const int lane  = threadIdx.x & 31;
  const int rhalf = (lane >> 4) << 3;
  const int col   = colBase + (lane & 15);
#pragma unroll
  for (int i = 0; i < 8; ++i) {
    int row = rowBase + i + rhalf;
    if (row < N) ZU[(size_t)row * 64 + col] = acc[i];
  }
}

// sign-split float atomic max (init must be -inf)
__device__ __forceinline__ void atomicMaxF(float* addr, float v) {
  if (v >= 0.f) atomicMax((int*)addr, __float_as_int(v));
  else          atomicMin((unsigned int*)addr, __float_as_uint(v));
}

__global__ void edge_logits_kernel(const float* __restrict__ ZU, const int* __restrict__ src,
                                   const int* __restrict__ dst, float* __restrict__ logit,
                                   float* __restrict__ segmax, int E) {
  int idx = blockIdx.x * blockDim.x + threadIdx.x;
  if (idx >= E * KH) return;
  int e = idx >> 2, k = idx & 3;
  int s = src[e], d = dst[e];
  const float4* zs = (const float4*)(ZU + (size_t)s * 64 + k * RDIM);
  const float4* zd = (const float4*)(ZU + (size_t)d * 64 + k * RDIM);
  float acc = 0.f;
#pragma unroll
  for (int q = 0; q < 4; ++q) {
    float4 a = zs[q], b = zd[q];
    acc += a.x * b.x + a.y * b.y + a.z * b.z + a.w * b.w;
  }
  float l = acc * 0.25f;                            // 1/sqrt(R), R=16
  logit[idx] = l;
  atomicMaxF(&segmax[d * KH + k], l);
}

__global__ void edge_exp_kernel(const int* __restrict__ dst, float* __restrict__ logit,
                                const float* __restrict__ segmax, float* __restrict__ segsum,
                                int E) {
  int idx = blockIdx.x * blockDim.x + threadIdx.x;
  if (idx >= E * KH) return;
  int e = idx >> 2, k = idx & 3;
  int d = dst[e];
  float ex = __expf(logit[idx] - segmax[d * KH + k]);
  logit[idx] = ex;                                  // reuse buffer as exp_l
  atomicAdd(&segsum[d * KH + k], ex);
}

__global__ void edge_agg_kernel(const float* __restrict__ ZU, const int* __restrict__ src,
                                const int* __restrict__ dst, const float* __restrict__ logit,
                                const float* __restrict__ segsum, float* __restrict__ agg,
                                int E) {
  int idx = blockIdx.x * blockDim.x + threadIdx.x;
  if (idx >= E * KH) return;
  int e = idx >> 2, k = idx & 3;
  int s = src[e], d = dst[e];
  float alpha = logit[idx] / (segsum[d * KH + k] + 1e-16f);
  const float4* zs = (const float4*)(ZU + (size_t)s * 64 + k * RDIM);
  float* ag = agg + (size_t)d * 64 + k * RDIM;
#pragma unroll
  for (int q = 0; q < 4; ++q) {
    float4 a = zs[q];
    atomicAdd(ag + q * 4 + 0, alpha * a.x);
    atomicAdd(ag + q * 4 + 1, alpha * a.y);
    atomicAdd(ag + q * 4 + 2, alpha * a.z);
    atomicAdd(ag + q * 4 + 3, alpha * a.w);
  }
}

// ---------------------------------------------------------------------------
// Z_half = 0.5*Z + 0.5*(agg (Nx64) @ Uw (64x256)). 8 waves x 2 col-tiles.
// ---------------------------------------------------------------------------
__global__ void zhalf_gemm_kernel(const float* __restrict__ agg, const float* __restrict__ Uw,
                                  const float* __restrict__ Z, float* __restrict__ Zhalf,
                                  int N) {
  __shared__ float aTile[16 * 68];
  const int rowBase = blockIdx.x * 16;
  stage_tile16(agg, 64, rowBase, N, 64, aTile, /*padCode 64dw*/5, 4);
  __syncthreads();
  const int wave  = threadIdx.x >> 5;              // 0..7
  const int lane  = threadIdx.x & 31;
  const int rhalf = (lane >> 4) << 3;
  const int lcol  = lane & 15;
#pragma unroll
  for (int t = 0; t < 2; ++t) {
    int colBase = (wave * 2 + t) * 16;
    v8f acc = {};
    acc = wmma_k_loop(aTile, 68, Uw + colBase, D_DIM, 64, acc);
#pragma unroll
    for (int i = 0; i < 8; ++i) {
      int row = rowBase + i + rhalf;
      if (row < N) {
        size_t o = (size_t)row * D_DIM + colBase + lcol;
        Zhalf[o] = 0.5f * Z[o] + 0.5f * acc[i];     // (1-C)*Z + C*ssa, C=0.5
      }
    }
  }
}

// ---------------------------------------------------------------------------
// Fused chain per 16-row tile, all intermediates in LDS:
//   DZ  = Zh @ Dmat^T   (B = DmatT, L2-resident)
//   res = Zh - DZ @ Dmat
//   out = relu(Zh + ETA*(res @ Dmat) - ETA*LAM)
// ---------------------------------------------------------------------------
__global__ void dict_chain_kernel(const float* __restrict__ Zhalf,
                                  const float* __restrict__ Dmat,
                                  const float* __restrict__ DmatT,
                                  float* __restrict__ out, int N) {
  __shared__ float zh[16 * 260];
  __shared__ float t1[16 * 260];
  __shared__ float t2[16 * 260];
  const int rowBase = blockIdx.x * 16;
  stage_tile16(Zhalf, D_DIM, rowBase, N, D_DIM, zh, 7, 4);
  __syncthreads();
  const int wave  = threadIdx.x >> 5;              // 0..7, 2 col-tiles each
  const int lane  = threadIdx.x & 31;
  const int rhalf = (lane >> 4) << 3;
  const int lcol  = lane & 15;

  // DZ = Zh @ Dmat^T
#pragma unroll
  for (int t = 0; t < 2; ++t) {
    int colBase = (wave * 2 + t) * 16;
    v8f acc = {};
    acc = wmma_k_loop(zh, 260, DmatT + colBase, D_DIM, D_DIM, acc);
#pragma unroll
    for (int i = 0; i < 8; ++i)
      t1[(i + rhalf) * 260 + colBase + lcol] = acc[i];
  }
  __syncthreads();

  // res = Zh - DZ @ Dmat
#pragma unroll
  for (int t = 0; t < 2; ++t) {
    int colBase = (wave * 2 + t) * 16;
    v8f acc = {};
    acc = wmma_k_loop(t1, 260, Dmat + colBase, D_DIM, D_DIM, acc);
#pragma unroll
    for (int i = 0; i < 8; ++i) {
      int r = i + rhalf;
      t2[r * 260 + colBase + lcol] = zh[r * 260 + colBase + lcol] - acc[i];
    }
  }
  __syncthreads();

  // out = relu(Zh + 0.5*(res @ Dmat) - 0.05)
#pragma unroll
  for (int t = 0; t < 2; ++t) {
    int colBase = (wave * 2 + t) * 16;
    v8f acc = {};
    acc = wmma_k_loop(t2, 260, Dmat + colBase, D_DIM, D_DIM, acc);
#pragma unroll
    for (int i = 0; i < 8; ++i) {
      int r   = i + rhalf;
      int row = rowBase + r;
      if (row < N) {
        float g = zh[r * 260 + colBase + lcol] + 0.5f * acc[i];
        out[(size_t)row * D_DIM + colBase + lcol] = fmaxf(g - 0.05f, 0.f);
      }
    }
  }
}

extern "C" void kernel_launch(void* const* d_in, const int* in_sizes, int n_in,
                              void* d_out, int out_size, void* d_ws, size_t ws_size,
                              hipStream_t stream) {
  const float* Z      = (const float*)d_in[0];
  const float* U      = (const float*)d_in[1];
  const float* Dmat   = (const float*)d_in[2];
  const float* head_w = (const float*)d_in[3];
  const int*   edges  = (const int*)d_in[4];

  const int N = in_sizes[0] / D_DIM;
  const int E = in_sizes[4] / 2;
  const int* srcIdx = edges;
  const int* dstIdx = edges + E;

  // workspace partition (each segment rounded to 64 floats = 256 B)
  float* p = (float*)d_ws;
  auto take = [&](size_t nElem) {
    float* r = p;
    p += (nElem + 63) & ~(size_t)63;
    return r;
  };
  float* ZU     = take((size_t)N * 64);
  float* logit  = take((size_t)E * KH);
  float* segmax = take((size_t)N * KH);
  float* segsum = take((size_t)N * KH);
  float* agg    = take((size_t)N * 64);
  float* Zhalf  = take((size_t)N * D_DIM);
  float* Ucat   = take((size_t)D_DIM * KH * RDIM);
  float* Uw     = take((size_t)KH * RDIM * D_DIM);
  float* DmatT  = take((size_t)D_DIM * D_DIM);

  const int tilesN     = (N + 15) / 16;
  const int edgeThr    = E * KH;
  const int edgeBlocks = (edgeThr + 255) / 256;

  pack_kernel<<<(D_DIM * D_DIM + 255) / 256, 256, 0, stream>>>(U, Dmat, head_w,
                                                               Ucat, Uw, DmatT);
  init_kernel<<<2048, 256, 0, stream>>>(segmax, segsum, agg, N);
  zu_gemm_kernel<<<tilesN, 128, 0, stream>>>(Z, Ucat, ZU, N);
  edge_logits_kernel<<<edgeBlocks, 256, 0, stream>>>(ZU, srcIdx, dstIdx, logit, segmax, E);
  edge_exp_kernel<<<edgeBlocks, 256, 0, stream>>>(dstIdx, logit, segmax, segsum, E);
  edge_agg_kernel<<<edgeBlocks, 256, 0, stream>>>(ZU, srcIdx, dstIdx, logit, segsum, agg, E);
  zhalf_gemm_kernel<<<tilesN, 256, 0, stream>>>(agg, Uw, Z, Zhalf, N);
  dict_chain_kernel<<<tilesN, 256, 0, stream>>>(Zhalf, Dmat, DmatT, (float*)d_out, N);
}